// cross_SS2D_21861383537444
// MI455X (gfx1250) — compile-verified
//
#include <hip/hip_runtime.h>
#include <cstdint>

#define B_  4
#define D_  96
#define L_  4096
#define N_  16
#define R_  6
#define K_  4
#define C_  38     // R + 2N
#define CP  48     // padded row length for x_dbl

typedef __attribute__((ext_vector_type(2))) float v2f;
typedef __attribute__((ext_vector_type(8))) float v8f;

__device__ __forceinline__ v8f wmma4(v2f a, v2f b, v8f c) {
  // D = A(16x4 f32) * B(4x16 f32) + C(16x16 f32), exact fp32 path on CDNA5
  return __builtin_amdgcn_wmma_f32_16x16x4_f32(
      /*neg_a=*/false, a, /*neg_b=*/false, b,
      /*c_mod=*/(short)0, c, /*reuse_a=*/false, /*reuse_b=*/false);
}

// ---------------------------------------------------------------------------
// Projection GEMM: x_dbl[b,k,l,c] = sum_d W[k,c,d] * src[b,d,col(l)]
// src column order is hw for k in {0,2}, wh (transposed) for k in {1,3}.
// One block = (stream, b, k, 128-column chunk); 8 waves, each owns a 16-col
// tile across the 3 row-tiles (c padded 38 -> 48). W staged in LDS.
// ---------------------------------------------------------------------------
__global__ __launch_bounds__(256) void proj_gemm_kernel(
    const float* __restrict__ x, const float* __restrict__ yv,
    const float* __restrict__ w_der, const float* __restrict__ w_cli,
    float* __restrict__ xdbl_der, float* __restrict__ xdbl_cli) {
  const int bid   = blockIdx.x;
  const int chunk = bid & 31;
  const int k     = (bid >> 5) & 3;
  const int b     = (bid >> 7) & 3;
  const int s     = bid >> 9;

  const float* src = s ? yv : x;
  const float* w   = (s ? w_cli : w_der) + k * C_ * D_;
  float* dst = (s ? xdbl_cli : xdbl_der) + ((size_t)(b * K_ + k)) * L_ * CP;

  __shared__ float Alds[CP * D_];   // 48*96*4 = 18 KB
  const int tid = threadIdx.x;
  for (int i = tid; i < CP * D_; i += 256) {
    const int c = i / D_, d = i - c * D_;
    Alds[i] = (c < C_) ? w[c * D_ + d] : 0.0f;
  }
  __syncthreads();

  const int lane = tid & 31;
  const int wv   = tid >> 5;
  const int n    = lane & 15;
  const int half = lane >> 4;
  const int l    = chunk * 128 + wv * 16 + n;
  const int col  = (k & 1) ? (((l & 63) << 6) | (l >> 6)) : l;  // wh -> hw map
  const float* sb = src + ((size_t)b * D_) * L_ + col;

  v8f a0 = {}, a1 = {}, a2 = {};
  for (int kk = 0; kk < 24; ++kk) {          // K = 96 in steps of 4
    const int d0 = kk * 4 + half * 2;
    v2f bf; bf.x = sb[(size_t)d0 * L_]; bf.y = sb[(size_t)(d0 + 1) * L_];
    const float* ap = &Alds[n * D_ + d0];
    v2f af0; af0.x = ap[0];           af0.y = ap[1];
    v2f af1; af1.x = ap[16 * D_];     af1.y = ap[16 * D_ + 1];
    v2f af2; af2.x = ap[32 * D_];     af2.y = ap[32 * D_ + 1];
    a0 = wmma4(af0, bf, a0);
    a1 = wmma4(af1, bf, a1);
    a2 = wmma4(af2, bf, a2);
  }

  // D layout: VGPR j -> row M = j + 8*half, col N = lane&15
  float* dcol = dst + (size_t)l * CP;
  const int cb = half * 8;
#pragma unroll
  for (int j = 0; j < 8; ++j) {
    dcol[cb + j]      = a0[j];
    dcol[16 + cb + j] = a1[j];
    const int c2 = 32 + cb + j;
    if (c2 < C_) dcol[c2] = a2[j];
  }
}

// ---------------------------------------------------------------------------
// Selective scan. Lane = (chain, n); 2 chains per wave; 16 chains per block.
// Block = (stream, b, k, 16-wide d chunk). Cross: C comes from other stream.
// ---------------------------------------------------------------------------
__global__ __launch_bounds__(256) void scan_kernel(
    const float* __restrict__ x, const float* __restrict__ yv,
    const float* __restrict__ dtw_der, const float* __restrict__ dtb_der,
    const float* __restrict__ Alog_der, const float* __restrict__ Ds_der,
    const float* __restrict__ dtw_cli, const float* __restrict__ dtb_cli,
    const float* __restrict__ Alog_cli, const float* __restrict__ Ds_cli,
    const float* __restrict__ xdbl_der, const float* __restrict__ xdbl_cli,
    float* __restrict__ yy_der, float* __restrict__ yy_cli) {
  const int bid    = blockIdx.x;
  const int dchunk = bid % 6;
  const int k      = (bid / 6) & 3;
  const int b      = (bid / 24) & 3;
  const int s      = bid / 96;

  const int tid  = threadIdx.x;
  const int n    = tid & 15;
  const int grp  = tid >> 4;            // chain within block (0..15)
  const int d    = dchunk * 16 + grp;
  const int kd   = k * D_ + d;

  const float* dtw = (s ? dtw_cli : dtw_der) + kd * R_;
  const float  dtb = (s ? dtb_cli : dtb_der)[kd];
  const float  Dsv = (s ? Ds_cli  : Ds_der )[kd];
  const float  Av  = -__expf((s ? Alog_cli : Alog_der)[kd * N_ + n]);

  const float* usrc = (s ? yv : x) + ((size_t)b * D_ + d) * L_;
  const float* xb = (s ? xdbl_cli : xdbl_der) + (size_t)(b * K_ + k) * L_ * CP;
  const float* xc = (s ? xdbl_der : xdbl_cli) + (size_t)(b * K_ + k) * L_ * CP;
  float* yy = (s ? yy_cli : yy_der) + ((size_t)b * D_ + d) * L_;

  const float w0 = dtw[0], w1 = dtw[1], w2 = dtw[2];
  const float w3 = dtw[3], w4 = dtw[4], w5 = dtw[5];

  float h = 0.0f;
  for (int p = 0; p < L_; ++p) {
    const int ps  = (k < 2) ? p : (L_ - 1 - p);        // flip for k>=2
    const int lhw = (k & 1) ? (((ps & 63) << 6) | (ps >> 6)) : ps;

    const float* row = xb + (size_t)ps * CP;
    // prefetch ~16 steps ahead along this chain's traversal
    {
      int pf = (k < 2) ? (ps + 16) : (ps - 16);
      pf = pf < 0 ? 0 : (pf > (L_ - 1) ? (L_ - 1) : pf);
      __builtin_prefetch(xb + (size_t)pf * CP, 0, 1);
    }

    const float dt = dtb + w0 * row[0] + w1 * row[1] + w2 * row[2]
                         + w3 * row[3] + w4 * row[4] + w5 * row[5];
    const float delta = (dt > 20.0f) ? dt : __logf(1.0f + __expf(dt));

    const float Bv = row[R_ + n];
    const float Cv = (xc + (size_t)ps * CP)[R_ + N_ + n];
    const float uv = usrc[lhw];

    h = h * __expf(delta * Av) + (delta * uv) * Bv;

    float val = h * Cv;                      // 16-lane dot with C
    val += __shfl_xor(val, 8, 16);
    val += __shfl_xor(val, 4, 16);
    val += __shfl_xor(val, 2, 16);
    val += __shfl_xor(val, 1, 16);
    if (n == 0) atomicAdd(&yy[lhw], val + uv * Dsv);   // 4-path overlay
  }
}

// ---------------------------------------------------------------------------
// Fused LayerNorm over D=96 per (stream,b,pixel); writes final [B,D,H,W].
// One wave per pixel, 3 d's per lane.
// ---------------------------------------------------------------------------
__global__ __launch_bounds__(256) void ln_kernel(
    const float* __restrict__ yy_der, const float* __restrict__ yy_cli,
    const float* __restrict__ s1, const float* __restrict__ b1,
    const float* __restrict__ s2, const float* __restrict__ b2,
    float* __restrict__ out) {
  const int gw   = blockIdx.x * 8 + (threadIdx.x >> 5);
  const int lane = threadIdx.x & 31;
  const int l    = gw & (L_ - 1);
  const int bq   = gw >> 12;
  const int b    = bq & 3;
  const int s    = bq >> 2;

  const float* yy = (s ? yy_cli : yy_der) + (size_t)b * D_ * L_ + l;
  float v[3];
#pragma unroll
  for (int j = 0; j < 3; ++j) v[j] = yy[(size_t)(lane + 32 * j) * L_];

  float sum = v[0] + v[1] + v[2];
  float sq  = v[0] * v[0] + v[1] * v[1] + v[2] * v[2];
#pragma unroll
  for (int o = 16; o >= 1; o >>= 1) {
    sum += __shfl_xor(sum, o, 32);
    sq  += __shfl_xor(sq,  o, 32);
  }
  const float m   = sum * (1.0f / D_);
  const float var = sq * (1.0f / D_) - m * m;
  const float rs  = rsqrtf(var + 1e-5f);

  const float* sc = s ? s2 : s1;
  const float* bi = s ? b2 : b1;
  float* ob = out + (size_t)s * B_ * D_ * L_ + (size_t)b * D_ * L_ + l;
#pragma unroll
  for (int j = 0; j < 3; ++j) {
    const int dd = lane + 32 * j;
    ob[(size_t)dd * L_] = (v[j] - m) * rs * sc[dd] + bi[dd];
  }
}

__global__ __launch_bounds__(256) void zero_kernel(float* __restrict__ p, int n) {
  const int i = blockIdx.x * 256 + threadIdx.x;
  if (i < n) p[i] = 0.0f;
}

// ---------------------------------------------------------------------------
extern "C" void kernel_launch(void* const* d_in, const int* in_sizes, int n_in,
                              void* d_out, int out_size, void* d_ws, size_t ws_size,
                              hipStream_t stream) {
  const float* x        = (const float*)d_in[0];
  const float* y        = (const float*)d_in[1];
  const float* w_der    = (const float*)d_in[2];
  const float* dtw_der  = (const float*)d_in[3];
  const float* dtb_der  = (const float*)d_in[4];
  const float* Alog_der = (const float*)d_in[5];
  const float* Ds_der   = (const float*)d_in[6];
  const float* w_cli    = (const float*)d_in[7];
  const float* dtw_cli  = (const float*)d_in[8];
  const float* dtb_cli  = (const float*)d_in[9];
  const float* Alog_cli = (const float*)d_in[10];
  const float* Ds_cli   = (const float*)d_in[11];
  const float* ln1s     = (const float*)d_in[12];
  const float* ln1b     = (const float*)d_in[13];
  const float* ln2s     = (const float*)d_in[14];
  const float* ln2b     = (const float*)d_in[15];

  float* ws = (float*)d_ws;
  const size_t XD = (size_t)B_ * K_ * L_ * CP;     // 3,145,728 floats
  const size_t YD = (size_t)B_ * D_ * L_;          // 1,572,864 floats
  float* xdbl_der = ws;
  float* xdbl_cli = ws + XD;
  float* yy_der   = ws + 2 * XD;
  float* yy_cli   = yy_der + YD;

  const int zn = (int)(2 * YD);
  zero_kernel<<<(zn + 255) / 256, 256, 0, stream>>>(yy_der, zn);

  proj_gemm_kernel<<<2 * B_ * K_ * 32, 256, 0, stream>>>(
      x, y, w_der, w_cli, xdbl_der, xdbl_cli);

  scan_kernel<<<2 * B_ * K_ * 6, 256, 0, stream>>>(
      x, y, dtw_der, dtb_der, Alog_der, Ds_der,
      dtw_cli, dtb_cli, Alog_cli, Ds_cli,
      xdbl_der, xdbl_cli, yy_der, yy_cli);

  ln_kernel<<<2 * B_ * L_ / 8, 256, 0, stream>>>(
      yy_der, yy_cli, ln1s, ln1b, ln2s, ln2b, (float*)d_out);
}